// GMMv2_12086037971141
// MI455X (gfx1250) — compile-verified
//
#include <hip/hip_runtime.h>

// =====================================================================
// GMM low-rank-MVN log-prob, B=1M, N=32, R=10, K=4  (MI455X / gfx1250)
//
// Memory floor: x = 128MB @ 23.3TB/s ~ 5.7us. All O(B) math is cast as
// fp32 WMMA (V_WMMA_F32_16X16X4_F32) so the VALU only does the tiny
// logsumexp epilogue. Two kernels: a one-block setup kernel that builds
// Cholesky / L^-1 / fragment tables in d_ws, then the WMMA main kernel.
// =====================================================================

typedef float v2f __attribute__((ext_vector_type(2)));
typedef float v8f __attribute__((ext_vector_type(8)));

#define NDIM 32
#define RNK  10
#define KC   4

// workspace (floats)
#define OFF_COLCONST 0            // [48]
#define OFF_CONSTA   48           // [4]
#define OFF_BFRAG    52           // [8 chunks][4 tiles][32 lanes][2]
#define OFF_GFRAG    (52 + 2048)  // [12 pairs][32 lanes][2]
#define PRE_FLOATS   (52 + 2048 + 768)

// stage-2 (chunk) table: tile k uses chunks covering cols 10k..10k+9
__device__ __constant__ int d_chk[12] = {0,1,2, 2,3,4, 5,6,7, 7,8,9};

// ---------------------------------------------------------------------
// Setup kernel: one block. Tiny (KB-scale) precompute.
// ---------------------------------------------------------------------
__global__ void __launch_bounds__(256)
gmm_setup_kernel(const float* __restrict__ w,    // [K]
                 const float* __restrict__ loc,  // [N,K]
                 const float* __restrict__ cf,   // [N,R,K]
                 const float* __restrict__ cd,   // [N,K]
                 float* __restrict__ pre) {
  __shared__ float sDinv[NDIM][KC];
  __shared__ float sC[KC][RNK][RNK];
  __shared__ float sLinv[KC][RNK][RNK];
  __shared__ float sc2[KC][RNK];
  __shared__ float sconstA[KC];

  const int t = threadIdx.x;

  if (t < NDIM * KC) {
    int n = t >> 2, k = t & 3;
    sDinv[n][k] = 1.0f / cd[n * KC + k];
  }
  __syncthreads();

  // C = I + W^T D^-1 W   (full symmetric fill)
  for (int idx = t; idx < KC * RNK * RNK; idx += blockDim.x) {
    int k = idx / (RNK * RNK);
    int r = (idx / RNK) % RNK;
    int s = idx % RNK;
    float acc = (r == s) ? 1.0f : 0.0f;
    for (int n = 0; n < NDIM; ++n)
      acc += cf[n * RNK * KC + r * KC + k] * sDinv[n][k] *
             cf[n * RNK * KC + s * KC + k];
    sC[k][r][s] = acc;
  }
  // c2[k][s] = sum_n Dinv*loc*cf
  if (t < KC * RNK) {
    int k = t / RNK, s = t % RNK;
    float a = 0.0f;
    for (int n = 0; n < NDIM; ++n)
      a += sDinv[n][k] * loc[n * KC + k] * cf[n * RNK * KC + s * KC + k];
    sc2[k][s] = a;
  }
  __syncthreads();

  // per-k Cholesky, inverse, constants (4 threads)
  if (t < KC) {
    const int k = t;
    float L[RNK][RNK];
    float Li[RNK][RNK];
    float logdetL = 0.0f;
    for (int i = 0; i < RNK; ++i) {
      float d = sC[k][i][i];
      for (int m = 0; m < i; ++m) d -= L[i][m] * L[i][m];
      d = sqrtf(d);
      L[i][i] = d;
      logdetL += logf(d);
      float inv = 1.0f / d;
      for (int j = i + 1; j < RNK; ++j) {
        float v = sC[k][j][i];
        for (int m = 0; m < i; ++m) v -= L[j][m] * L[i][m];
        L[j][i] = v * inv;
      }
    }
    for (int j = 0; j < RNK; ++j) {
      Li[j][j] = 1.0f / L[j][j];
      for (int i = j + 1; i < RNK; ++i) {
        float s2 = 0.0f;
        for (int m = j; m < i; ++m) s2 += L[i][m] * Li[m][j];
        Li[i][j] = -s2 / L[i][i];
      }
    }
    for (int i = 0; i < RNK; ++i)
      for (int j = 0; j < RNK; ++j)
        sLinv[k][i][j] = (i >= j) ? Li[i][j] : 0.0f;

    // log-softmax of mixture weights (K=4)
    float w0 = w[0], w1 = w[1], w2 = w[2], w3 = w[3];
    float m = fmaxf(fmaxf(w0, w1), fmaxf(w2, w3));
    float lse = m + logf(expf(w0 - m) + expf(w1 - m) + expf(w2 - m) + expf(w3 - m));
    float lw = w[k] - lse;

    float logdetD = 0.0f, c1 = 0.0f;
    for (int n = 0; n < NDIM; ++n) {
      logdetD += logf(cd[n * KC + k]);
      float l = loc[n * KC + k];
      c1 += sDinv[n][k] * l * l;
    }
    sconstA[k] = lw - 0.5f * ((float)NDIM * logf(6.283185307179586f) +
                              logdetD + 2.0f * logdetL + c1);
  }
  __syncthreads();

  // ---- emit tables ----
  if (t < 48) {
    float v = 0.0f;
    if (t < 40) { int k = t / RNK, s = t % RNK; v = sc2[k][s]; }
    pre[OFF_COLCONST + t] = v;
  }
  if (t < KC) pre[OFF_CONSTA + t] = sconstA[t];

  // Stage-1 B fragments. WMMA f32 16x16x4 fragment layout assumption
  // (mirrors documented A layout): elem i, lane half h -> K = i + 2h,
  // lane%16 -> N (column).
  for (int u = t; u < 8 * 4 * 32; u += blockDim.x) {
    int c = u >> 7, t2 = (u >> 5) & 3, lane = u & 31;
    int h = lane >> 4, ccl = lane & 15;
#pragma unroll
    for (int i = 0; i < 2; ++i) {
      int n = 4 * c + i + 2 * h;  // global reduction index (x column)
      float v = 0.0f;
      if (t2 < 3) {               // x-pass tiles: cols = 16*t2 + ccl
        int col = t2 * 16 + ccl;
        if (col < 40) {           // A columns: col = 10k + s  -> WD[n,s,k]
          int k = col / RNK, s = col % RNK;
          v = cf[n * RNK * KC + s * KC + k] * sDinv[n][k];
        } else if (col < 44) {    // s1 columns: Dinv*loc
          int k = col - 40;
          v = sDinv[n][k] * loc[n * KC + k];
        }
      } else {                    // x^2-pass tile: cols 0..3 = Dinv
        if (ccl < KC) v = sDinv[n][ccl];
      }
      pre[OFF_BFRAG + u * 2 + i] = v;
    }
  }

  // Stage-2 G fragments: y[(k,r)] = sum_s Linv_k[r][s] * T[.,10k+s]
  for (int u = t; u < 12 * 32; u += blockDim.x) {
    int p = u >> 5, lane = u & 31;
    int h = lane >> 4, r = lane & 15;
    int kt = p / 3, c = d_chk[p];
#pragma unroll
    for (int i = 0; i < 2; ++i) {
      int row = 4 * c + i + 2 * h;  // K index = T column
      float v = 0.0f;
      if (row < 40) {
        int kp = row / RNK, s = row % RNK;
        if (kp == kt && r < RNK) v = sLinv[kt][r][s];
      }
      pre[OFF_GFRAG + u * 2 + i] = v;
    }
  }
}

// ---------------------------------------------------------------------
// Main kernel: one wave handles 16 samples per iteration.
// ---------------------------------------------------------------------
#define TSTRIDE 52  // 48 cols + pad; 52*r mod 64 distinct for r=0..15

__global__ void __launch_bounds__(256)
gmm_main_kernel(const float* __restrict__ x,
                const float* __restrict__ pre,
                float* __restrict__ out,
                int ntiles) {
  __shared__ float smem[8 * 16 * TSTRIDE];

  const int lane = threadIdx.x & 31;
  const int wib  = threadIdx.x >> 5;
  const int gw   = blockIdx.x * (blockDim.x >> 5) + wib;
  const int nw   = gridDim.x * (blockDim.x >> 5);
  const int h    = lane >> 4;   // lane half
  const int c    = lane & 15;   // column / row-in-tile index
  float* T = &smem[wib * 16 * TSTRIDE];

  // persistent per-lane fragment registers (loaded once)
  v2f bf[8][3], bf2[8], gf[12];
  const v2f* bptr = (const v2f*)(pre + OFF_BFRAG);
#pragma unroll
  for (int ch = 0; ch < 8; ++ch) {
#pragma unroll
    for (int t2 = 0; t2 < 3; ++t2) bf[ch][t2] = bptr[(ch * 4 + t2) * 32 + lane];
    bf2[ch] = bptr[(ch * 4 + 3) * 32 + lane];
  }
  const v2f* gptr = (const v2f*)(pre + OFF_GFRAG);
#pragma unroll
  for (int p = 0; p < 12; ++p) gf[p] = gptr[p * 32 + lane];

  const float cc0 = pre[OFF_COLCONST + c];
  const float cc1 = pre[OFF_COLCONST + 16 + c];
  const float cc2 = pre[OFF_COLCONST + 32 + c];
  const float cA0 = pre[OFF_CONSTA + 0];
  const float cA1 = pre[OFF_CONSTA + 1];
  const float cA2 = pre[OFF_CONSTA + 2];
  const float cA3 = pre[OFF_CONSTA + 3];

  for (int tile = gw; tile < ntiles; tile += nw) {
    const float* xrow = x + (size_t)(tile * 16 + c) * NDIM;  // A row M = c

    v8f acc0 = {0,0,0,0,0,0,0,0};
    v8f acc1 = {0,0,0,0,0,0,0,0};
    v8f acc2 = {0,0,0,0,0,0,0,0};
    v8f acc3 = {0,0,0,0,0,0,0,0};

    // ---- stage 1: T[16x48] = [x | x^2] x [WD | Dinv*loc | Dinv] ----
#pragma unroll
    for (int ch = 0; ch < 8; ++ch) {
      v2f xa = *(const v2f*)(xrow + 4 * ch + 2 * h);  // K = i + 2h
      v2f xs = xa * xa;
      acc0 = __builtin_amdgcn_wmma_f32_16x16x4_f32(false, xa, false, bf[ch][0],
                                                   (short)0, acc0, false, false);
      acc1 = __builtin_amdgcn_wmma_f32_16x16x4_f32(false, xa, false, bf[ch][1],
                                                   (short)0, acc1, false, false);
      acc2 = __builtin_amdgcn_wmma_f32_16x16x4_f32(false, xa, false, bf[ch][2],
                                                   (short)0, acc2, false, false);
      acc3 = __builtin_amdgcn_wmma_f32_16x16x4_f32(false, xs, false, bf2[ch],
                                                   (short)0, acc3, false, false);
    }

    // subtract per-column constants (A = x.WD - c2; zero for s1/s2 cols)
#pragma unroll
    for (int j = 0; j < 8; ++j) {
      acc0[j] -= cc0;
      acc1[j] -= cc1;
      acc2[j] -= cc2;
    }

    // ---- spill T to LDS (D layout: vgpr j -> row j + 8h, lane -> col) ----
#pragma unroll
    for (int j = 0; j < 8; ++j) {
      int row = j + 8 * h;
      T[row * TSTRIDE + c]      = acc0[j];
      T[row * TSTRIDE + 16 + c] = acc1[j];
      T[row * TSTRIDE + 32 + c] = acc2[j];  // cols 44..47 are zeros here
    }
    if (c < KC) {
#pragma unroll
      for (int j = 0; j < 8; ++j) {
        int row = j + 8 * h;
        T[row * TSTRIDE + 44 + c] = acc3[j];  // s2 overwrites the zeros
      }
    }
    // per-wave LDS is in-order; just stop the compiler reordering.
    asm volatile("" ::: "memory");

    // ---- stage 2: y = T x G  (G = blockdiag(Linv^T), 3 chunks/block) ----
    v2f aT[10];
#pragma unroll
    for (int ch = 0; ch < 10; ++ch)
      aT[ch] = *(const v2f*)&T[c * TSTRIDE + 4 * ch + 2 * h];  // A row M=c

    v8f y0 = {0,0,0,0,0,0,0,0};
    v8f y1 = {0,0,0,0,0,0,0,0};
    v8f y2 = {0,0,0,0,0,0,0,0};
    v8f y3 = {0,0,0,0,0,0,0,0};
    y0 = __builtin_amdgcn_wmma_f32_16x16x4_f32(false, aT[0], false, gf[0],  (short)0, y0, false, false);
    y0 = __builtin_amdgcn_wmma_f32_16x16x4_f32(false, aT[1], false, gf[1],  (short)0, y0, false, false);
    y0 = __builtin_amdgcn_wmma_f32_16x16x4_f32(false, aT[2], false, gf[2],  (short)0, y0, false, false);
    y1 = __builtin_amdgcn_wmma_f32_16x16x4_f32(false, aT[2], false, gf[3],  (short)0, y1, false, false);
    y1 = __builtin_amdgcn_wmma_f32_16x16x4_f32(false, aT[3], false, gf[4],  (short)0, y1, false, false);
    y1 = __builtin_amdgcn_wmma_f32_16x16x4_f32(false, aT[4], false, gf[5],  (short)0, y1, false, false);
    y2 = __builtin_amdgcn_wmma_f32_16x16x4_f32(false, aT[5], false, gf[6],  (short)0, y2, false, false);
    y2 = __builtin_amdgcn_wmma_f32_16x16x4_f32(false, aT[6], false, gf[7],  (short)0, y2, false, false);
    y2 = __builtin_amdgcn_wmma_f32_16x16x4_f32(false, aT[7], false, gf[8],  (short)0, y2, false, false);
    y3 = __builtin_amdgcn_wmma_f32_16x16x4_f32(false, aT[7], false, gf[9],  (short)0, y3, false, false);
    y3 = __builtin_amdgcn_wmma_f32_16x16x4_f32(false, aT[8], false, gf[10], (short)0, y3, false, false);
    y3 = __builtin_amdgcn_wmma_f32_16x16x4_f32(false, aT[9], false, gf[11], (short)0, y3, false, false);

    // ---- quad_k = |y_k|^2 : square + 16-lane xor-shuffle reduce ----
    // (columns r>=10 are exact zeros since G is zero there)
    v8f q0 = y0 * y0, q1 = y1 * y1, q2 = y2 * y2, q3 = y3 * y3;
#pragma unroll
    for (int m = 1; m < 16; m <<= 1) {
#pragma unroll
      for (int j = 0; j < 8; ++j) {
        q0[j] += __shfl_xor(q0[j], m, 32);
        q1[j] += __shfl_xor(q1[j], m, 32);
        q2[j] += __shfl_xor(q2[j], m, 32);
        q3[j] += __shfl_xor(q3[j], m, 32);
      }
    }

    // ---- logsumexp epilogue:  val_k = constA_k + s1 - s2/2 + quad/2 ----
#pragma unroll
    for (int j = 0; j < 8; ++j) {
      int row = j + 8 * h;
      const float* tr = &T[row * TSTRIDE];
      float4 s1v = *(const float4*)&tr[40];
      float4 s2v = *(const float4*)&tr[44];
      float v0 = cA0 + s1v.x - 0.5f * s2v.x + 0.5f * q0[j];
      float v1 = cA1 + s1v.y - 0.5f * s2v.y + 0.5f * q1[j];
      float v2 = cA2 + s1v.z - 0.5f * s2v.z + 0.5f * q2[j];
      float v3 = cA3 + s1v.w - 0.5f * s2v.w + 0.5f * q3[j];
      float mx  = fmaxf(fmaxf(v0, v1), fmaxf(v2, v3));
      float sum = __expf(v0 - mx) + __expf(v1 - mx) +
                  __expf(v2 - mx) + __expf(v3 - mx);
      float res = mx + __logf(sum);
      if (c == j) out[tile * 16 + row] = res;  // lanes j / j+16 cover 16 rows
    }
  }
}

// ---------------------------------------------------------------------
extern "C" void kernel_launch(void* const* d_in, const int* in_sizes, int n_in,
                              void* d_out, int out_size, void* d_ws, size_t ws_size,
                              hipStream_t stream) {
  (void)n_in; (void)out_size; (void)ws_size;
  const float* x   = (const float*)d_in[0];
  const float* w   = (const float*)d_in[1];
  const float* loc = (const float*)d_in[2];
  const float* cf  = (const float*)d_in[3];
  const float* cd  = (const float*)d_in[4];
  float* out = (float*)d_out;
  float* pre = (float*)d_ws;  // needs PRE_FLOATS*4 = ~11.5 KB of scratch

  const int Bn = in_sizes[0] / NDIM;
  const int ntiles = Bn / 16;
  if (ntiles <= 0) return;

  gmm_setup_kernel<<<1, 256, 0, stream>>>(w, loc, cf, cd, pre);

  int blocks = (ntiles + 63) / 64;   // 8 waves/block, ~8 tiles/wave
  if (blocks > 1024) blocks = 1024;
  if (blocks < 1)    blocks = 1;
  gmm_main_kernel<<<blocks, 256, 0, stream>>>(x, pre, out, ntiles);
}